// GPT2MQAttention_76081050681809
// MI455X (gfx1250) — compile-verified
//
#include <hip/hip_runtime.h>
#include <hip/hip_bf16.h>
#include <math.h>

typedef __attribute__((ext_vector_type(2))) float v2f;
typedef __attribute__((ext_vector_type(8))) float v8f;

constexpr int kE    = 4096;
constexpr int kH    = 32;
constexpr int kD    = 128;
constexpr int kPAST = 8191;
constexpr int kK    = 8192;   // PAST + 1

// ---------------------------------------------------------------------------
// Kernel 1: fused q/kv GEMV.  Each block: 64 output columns (16 groups of 4,
// float4/b128 streaming loads), 16-way K split, deterministic LDS reduction.
// Blocks 0..63 -> q (q_w, stride 4096); blocks 64..67 -> kv (kv_w, stride 256).
// ---------------------------------------------------------------------------
__global__ __launch_bounds__(256) void qkv_gemv(
    const float* __restrict__ h,
    const float* __restrict__ qw,  const float* __restrict__ qb,
    const float* __restrict__ kvw, const float* __restrict__ kvb,
    float* __restrict__ qout, float* __restrict__ kvout)
{
    __shared__ float  sh[kE];
    __shared__ float4 sred[256];
    const int t = threadIdx.x;

    const float4* h4  = (const float4*)h;
    float4*       sh4 = (float4*)sh;
    for (int i = t; i < kE / 4; i += 256) sh4[i] = h4[i];
    __syncthreads();

    const int cg  = t & 15;     // column group (4 consecutive cols)
    const int seg = t >> 4;     // 0..15, each owns 256 rows

    const float* W; const float* bias; float* dst; int stride; int jb;
    if (blockIdx.x < 64) { W = qw;  stride = kE;     bias = qb;  dst = qout;
                           jb = blockIdx.x * 64; }
    else                 { W = kvw; stride = 2 * kD; bias = kvb; dst = kvout;
                           jb = (blockIdx.x - 64) * 64; }
    const int col = jb + cg * 4;

    float ax = 0.f, ay = 0.f, az = 0.f, aw = 0.f;
    const int i0 = seg * 256;
    for (int ii = i0; ii < i0 + 256; ii += 32) {
        __builtin_prefetch(&W[(size_t)(ii + 32) * stride + col], 0, 1);
        #pragma unroll 8
        for (int i = ii; i < ii + 32; ++i) {
            const float4 w  = *(const float4*)(W + (size_t)i * stride + col);
            const float  hv = sh[i];
            ax = fmaf(hv, w.x, ax); ay = fmaf(hv, w.y, ay);
            az = fmaf(hv, w.z, az); aw = fmaf(hv, w.w, aw);
        }
    }
    sred[t] = make_float4(ax, ay, az, aw);
    __syncthreads();
    if (t < 16) {
        float sx = 0.f, sy = 0.f, sz = 0.f, sw = 0.f;
        #pragma unroll
        for (int sg = 0; sg < 16; ++sg) {
            const float4 v = sred[sg * 16 + t];
            sx += v.x; sy += v.y; sz += v.z; sw += v.w;
        }
        const int c = jb + t * 4;
        dst[c + 0] = sx + bias[c + 0];
        dst[c + 1] = sy + bias[c + 1];
        dst[c + 2] = sz + bias[c + 2];
        dst[c + 3] = sw + bias[c + 3];
    }
}

// ---------------------------------------------------------------------------
// Kernel 2: scores = (Q[32x128] @ Kfull[128x8192]) * 1/sqrt(D) + mask
// One wave per 16x16 tile, V_WMMA_F32_16X16X4_F32 over K=128.
// The lane-divergent boundary (key col 8191 = new key) is hoisted out of the
// loop as a per-lane base pointer + stride -> branchless inner loop.
// ---------------------------------------------------------------------------
__global__ __launch_bounds__(32) void scores_wmma(
    const float* __restrict__ q,  const float* __restrict__ pk,
    const float* __restrict__ kv, const float* __restrict__ am,
    float* __restrict__ scores)
{
    const int lane  = threadIdx.x;
    const int half  = lane >> 4;
    const int nl    = lane & 15;
    const int ntile = blockIdx.x;      // 0..511 (key positions)
    const int mtile = blockIdx.y;      // 0..1   (head tiles)
    const int kidx  = ntile * 16 + nl; // key position 0..8191

    const float* qrow = q + (size_t)(mtile * 16 + nl) * kD;

    // per-lane B source: column kidx of past_key (row stride PAST floats),
    // or the fresh key vector kv[0:128] (stride 1) for kidx == 8191
    const float* bptr;  size_t bstr;
    if (kidx < kPAST) { bptr = pk + kidx; bstr = (size_t)kPAST; }
    else              { bptr = kv;        bstr = 1;             }

    v8f c = {};
    for (int kb = 0; kb < kD; kb += 4) {
        const int r0 = kb + 2 * half;
        v2f a, b;
        a.x = qrow[r0];
        a.y = qrow[r0 + 1];
        b.x = bptr[(size_t)r0 * bstr];
        b.y = bptr[(size_t)(r0 + 1) * bstr];
        c = __builtin_amdgcn_wmma_f32_16x16x4_f32(
                false, a, false, b, (short)0, c, false, false);
    }

    const float scale = 0.08838834764831845f;   // 1/sqrt(128)
    const float maskv = am[kidx];
    #pragma unroll
    for (int v = 0; v < 8; ++v) {
        const int M = v + 8 * half;             // C/D layout
        scores[(size_t)(mtile * 16 + M) * kK + kidx] = c[v] * scale + maskv;
    }
}

// ---------------------------------------------------------------------------
// Kernel 3: row softmax over 8192, one block per head (exact, max-subtracted).
// Block 0 additionally copies (key_new, value_new) to the output tail.
// ---------------------------------------------------------------------------
__global__ __launch_bounds__(256) void softmax_row(
    float* __restrict__ s, const float* __restrict__ kv,
    float* __restrict__ outtail)
{
    __shared__ float red[256];
    const int hrow = blockIdx.x, t = threadIdx.x;
    float* row = s + (size_t)hrow * kK;

    if (hrow == 0) outtail[t] = kv[t];     // 128 key + 128 value

    float vals[32];
    float mx = -3.402823466e+38f;
    #pragma unroll
    for (int i = 0; i < 32; ++i) { vals[i] = row[t + 256 * i]; mx = fmaxf(mx, vals[i]); }
    red[t] = mx; __syncthreads();
    for (int o = 128; o > 0; o >>= 1) {
        if (t < o) red[t] = fmaxf(red[t], red[t + o]);
        __syncthreads();
    }
    mx = red[0]; __syncthreads();

    float sum = 0.f;
    #pragma unroll
    for (int i = 0; i < 32; ++i) { vals[i] = __expf(vals[i] - mx); sum += vals[i]; }
    red[t] = sum; __syncthreads();
    for (int o = 128; o > 0; o >>= 1) {
        if (t < o) red[t] += red[t + o];
        __syncthreads();
    }
    const float inv = 1.f / red[0];
    #pragma unroll
    for (int i = 0; i < 32; ++i) row[t + 256 * i] = vals[i] * inv;
}

// ---------------------------------------------------------------------------
// Kernel 4: ctx = W[32x8192] @ Vfull[8192x128], K-split 16 ways into partials.
// Main loop is branchless over past_value; the single K-step touching the new
// value row (kb == 8188, row 8191) runs as a one-time epilogue.
// ---------------------------------------------------------------------------
__global__ __launch_bounds__(32) void attnout_wmma(
    const float* __restrict__ w,  const float* __restrict__ pv,
    const float* __restrict__ kv, float* __restrict__ part)
{
    const int lane  = threadIdx.x;
    const int half  = lane >> 4;
    const int nl    = lane & 15;
    const int kc    = blockIdx.x;   // 0..15 (512-wide K chunk)
    const int mtile = blockIdx.y;   // 0..1
    const int ntile = blockIdx.z;   // 0..7
    const int nidx  = ntile * 16 + nl;   // d index 0..127

    const float* wrow = w + (size_t)(mtile * 16 + nl) * kK;

    v8f c = {};
    const int k0    = kc * 512;
    const int kend  = k0 + 512;
    const int kmain = (kend > 8188) ? 8188 : kend;   // last step is special

    for (int kb = k0; kb < kmain; kb += 4) {
        const int r0 = kb + 2 * half;
        v2f a, b;
        a.x = wrow[r0];
        a.y = wrow[r0 + 1];
        b.x = pv[(size_t)r0 * kD + nidx];
        b.y = pv[(size_t)(r0 + 1) * kD + nidx];
        c = __builtin_amdgcn_wmma_f32_16x16x4_f32(
                false, a, false, b, (short)0, c, false, false);
    }
    if (kend > 8188) {              // kb == 8188: rows 8188..8190 + new row 8191
        const int r0 = 8188 + 2 * half;
        v2f a, b;
        a.x = wrow[r0];
        a.y = wrow[r0 + 1];
        b.x = pv[(size_t)r0 * kD + nidx];
        b.y = half ? kv[kD + nidx] : pv[(size_t)(r0 + 1) * kD + nidx];
        c = __builtin_amdgcn_wmma_f32_16x16x4_f32(
                false, a, false, b, (short)0, c, false, false);
    }

    #pragma unroll
    for (int v = 0; v < 8; ++v) {
        const int M = v + 8 * half;
        part[(size_t)kc * (kH * kD) + (size_t)(mtile * 16 + M) * kD + nidx] = c[v];
    }
}

// ---------------------------------------------------------------------------
// Kernel 5: out = ctx @ proj_w + proj_b.  The 16-way K-chunk reduction of the
// context partials is fused into the LDS staging loop (L2-hot, fixed order).
// Same float4/b128 streaming GEMV scheme as kernel 1.  64 blocks.
// ---------------------------------------------------------------------------
__global__ __launch_bounds__(256) void proj_gemv(
    const float* __restrict__ part, const float* __restrict__ W,
    const float* __restrict__ b, float* __restrict__ out)
{
    __shared__ float  sh[kE];
    __shared__ float4 sred[256];
    const int t = threadIdx.x;

    for (int i = t; i < kE; i += 256) {
        float s = 0.f;
        #pragma unroll
        for (int c = 0; c < 16; ++c) s += part[(size_t)c * kE + i];
        sh[i] = s;
    }
    __syncthreads();

    const int cg  = t & 15;
    const int seg = t >> 4;
    const int jb  = blockIdx.x * 64;
    const int col = jb + cg * 4;

    float ax = 0.f, ay = 0.f, az = 0.f, aw = 0.f;
    const int i0 = seg * 256;
    for (int ii = i0; ii < i0 + 256; ii += 32) {
        __builtin_prefetch(&W[(size_t)(ii + 32) * kE + col], 0, 1);
        #pragma unroll 8
        for (int i = ii; i < ii + 32; ++i) {
            const float4 w  = *(const float4*)(W + (size_t)i * kE + col);
            const float  hv = sh[i];
            ax = fmaf(hv, w.x, ax); ay = fmaf(hv, w.y, ay);
            az = fmaf(hv, w.z, az); aw = fmaf(hv, w.w, aw);
        }
    }
    sred[t] = make_float4(ax, ay, az, aw);
    __syncthreads();
    if (t < 16) {
        float sx = 0.f, sy = 0.f, sz = 0.f, sw = 0.f;
        #pragma unroll
        for (int sg = 0; sg < 16; ++sg) {
            const float4 v = sred[sg * 16 + t];
            sx += v.x; sy += v.y; sz += v.z; sw += v.w;
        }
        const int c = jb + t * 4;
        out[c + 0] = sx + b[c + 0];
        out[c + 1] = sy + b[c + 1];
        out[c + 2] = sz + b[c + 2];
        out[c + 3] = sw + b[c + 3];
    }
}

// ---------------------------------------------------------------------------
extern "C" void kernel_launch(void* const* d_in, const int* in_sizes, int n_in,
                              void* d_out, int out_size, void* d_ws, size_t ws_size,
                              hipStream_t stream)
{
    const float* hs  = (const float*)d_in[0];   // (1,1,4096)
    const float* pk  = (const float*)d_in[1];   // (1,128,8191)
    const float* pv  = (const float*)d_in[2];   // (1,8191,128)
    const float* am  = (const float*)d_in[3];   // (1,1,1,8192)
    const float* qw  = (const float*)d_in[4];   // (4096,4096)
    const float* qb  = (const float*)d_in[5];   // (4096)
    const float* kvw = (const float*)d_in[6];   // (4096,256)
    const float* kvb = (const float*)d_in[7];   // (256)
    const float* pw  = (const float*)d_in[8];   // (4096,4096)
    const float* pb  = (const float*)d_in[9];   // (4096)
    float* out = (float*)d_out;                 // 4096 + 128 + 128

    float* ws   = (float*)d_ws;
    float* qv   = ws;              // 4096
    float* kvv  = ws + 4096;       // 256
    float* sc   = ws + 4352;       // 32*8192 = 262144
    float* part = ws + 266496;     // 16*4096 = 65536   (total ~1.27 MB)

    qkv_gemv    <<<68, 256, 0, stream>>>(hs, qw, qb, kvw, kvb, qv, kvv);
    scores_wmma <<<dim3(512, 2), 32, 0, stream>>>(qv, pk, kvv, am, sc);
    softmax_row <<<32, 256, 0, stream>>>(sc, kvv, out + kE);
    attnout_wmma<<<dim3(16, 2, 8), 32, 0, stream>>>(sc, pv, kvv, part);
    proj_gemv   <<<64, 256, 0, stream>>>(part, pw, pb, out);
}